// EnhancedTherapeuticGNN_65189013618817
// MI455X (gfx1250) — compile-verified
//
#include <hip/hip_runtime.h>
#include <hip/hip_bf16.h>

// ---------------------------------------------------------------------------
// EnhancedTherapeuticGNN on gfx1250 (MI455X): 2x GAT(128->128) + 2 heads.
// Dense GEMMs use V_WMMA_F32_16X16X4_F32 (full f32 precision).
// Edge softmax/aggregation uses f32 global atomics (hits 192MB L2).
// ---------------------------------------------------------------------------

typedef float v2f __attribute__((ext_vector_type(2)));
typedef float v8f __attribute__((ext_vector_type(8)));

#define HIDC 128
#define NEG_SLOPE 0.2f

// ---------------------------------------------------------------------------
// GEMM: H[M,128] = A[M,128] @ W^T, W stored row-major [128(out),128(in)].
// One block = 16-row M tile; 8 waves each own a 16-col N tile; K loop of 4.
// WMMA f32 16x16x4 fragment layout (ISA 7.12.2):
//   A: lane<16 -> (M=lane, K=k0+{0,1}); lane>=16 -> (M=lane-16, K=k0+{2,3})
//   B: lane<16 -> (N=lane, K=k0+{0,1}); lane>=16 -> (N=lane-16, K=k0+{2,3})
//   D: vgpr r  -> lane<16: (M=r, N=lane); lane>=16: (M=r+8, N=lane-16)
// ---------------------------------------------------------------------------
__global__ __launch_bounds__(256) void gemm128_wmma_f32(
    const float* __restrict__ A, const float* __restrict__ W,
    float* __restrict__ H, int M)
{
    const int wave = threadIdx.x >> 5;      // 0..7 -> N tile
    const int lane = threadIdx.x & 31;
    const int m0   = blockIdx.x * 16;
    const int n0   = wave * 16;
    const int r    = lane & 15;
    const int kh   = (lane >> 4) * 2;       // 0 or 2

    int arow = m0 + r; if (arow >= M) arow = M - 1;   // clamp (M%16==0 here)
    const float* Ap = A + (size_t)arow * HIDC + kh;
    const float* Wp = W + (size_t)(n0 + r) * HIDC + kh;

    v8f acc = {};
#pragma unroll
    for (int k0 = 0; k0 < HIDC; k0 += 4) {
        v2f a = *(const v2f*)(Ap + k0);
        v2f b = *(const v2f*)(Wp + k0);
        acc = __builtin_amdgcn_wmma_f32_16x16x4_f32(
            /*neg_a=*/false, a, /*neg_b=*/false, b,
            /*c_mod=*/(short)0, acc, /*reuse_a=*/false, /*reuse_b=*/false);
    }

    const int n     = n0 + r;
    const int mbase = m0 + ((lane >> 4) << 3);
#pragma unroll
    for (int i = 0; i < 8; ++i) {
        int m = mbase + i;
        if (m < M) H[(size_t)m * HIDC + n] = acc[i];
    }
}

// ---------------------------------------------------------------------------
// Per-node attention logits: alpha_s = h . a_src ; alpha_d = h . a_dst
// One wave per node, float4 per lane, butterfly reduce (wave32).
// ---------------------------------------------------------------------------
__global__ __launch_bounds__(256) void node_alpha(
    const float* __restrict__ H, const float* __restrict__ a_src,
    const float* __restrict__ a_dst, float* __restrict__ as,
    float* __restrict__ ad, int n)
{
    int wid  = (int)((blockIdx.x * (size_t)blockDim.x + threadIdx.x) >> 5);
    int lane = threadIdx.x & 31;
    if (wid >= n) return;
    float4 h4 = *(const float4*)(H + (size_t)wid * HIDC + lane * 4);
    float4 s4 = *(const float4*)(a_src + lane * 4);
    float4 d4 = *(const float4*)(a_dst + lane * 4);
    float s = h4.x * s4.x + h4.y * s4.y + h4.z * s4.z + h4.w * s4.w;
    float d = h4.x * d4.x + h4.y * d4.y + h4.z * d4.z + h4.w * d4.w;
#pragma unroll
    for (int off = 16; off > 0; off >>= 1) {
        s += __shfl_xor(s, off, 32);
        d += __shfl_xor(d, off, 32);
    }
    if (lane == 0) { as[wid] = s; ad[wid] = d; }
}

// ---------------------------------------------------------------------------
// Init per-layer scratch: accum=0, emax=-inf, denom=0
// ---------------------------------------------------------------------------
__global__ void init_layer(float* __restrict__ accum, float* __restrict__ emax,
                           float* __restrict__ denom, int n)
{
    size_t i = blockIdx.x * (size_t)blockDim.x + threadIdx.x;
    size_t tot = (size_t)n * HIDC;
    if (i < tot) accum[i] = 0.f;
    if (i < (size_t)n) { emax[i] = -__builtin_inff(); denom[i] = 0.f; }
}

__device__ __forceinline__ void atomicMaxFloat(float* addr, float val)
{
    // valid with -inf init: positive path via signed-int max, negative via uint min
    if (val >= 0.f) atomicMax((int*)addr, __float_as_int(val));
    else            atomicMin((unsigned int*)addr, __float_as_uint(val));
}

__device__ __forceinline__ void edge_endpoints(const long long* __restrict__ ei,
                                               int e, int E, int& src, int& dst)
{
    if (e < E) { src = (int)ei[e]; dst = (int)ei[(size_t)E + e]; }
    else       { src = dst = e - E; }          // self loops appended
}

// Pass A: e = leaky_relu(as[src] + ad[dst]); segment max over dst
__global__ void edge_logit_max(const long long* __restrict__ ei,
                               const float* __restrict__ as,
                               const float* __restrict__ ad,
                               float* __restrict__ evals,
                               float* __restrict__ emax, int Etot, int E)
{
    int e = (int)(blockIdx.x * (size_t)blockDim.x + threadIdx.x);
    if (e >= Etot) return;
    int src, dst; edge_endpoints(ei, e, E, src, dst);
    float v = as[src] + ad[dst];
    v = (v > 0.f) ? v : NEG_SLOPE * v;
    evals[e] = v;
    atomicMaxFloat(&emax[dst], v);
}

// Pass B: w = exp(e - emax[dst]); segment sum over dst (in-place e -> w)
__global__ void edge_exp_sum(const long long* __restrict__ ei,
                             float* __restrict__ evals,
                             const float* __restrict__ emax,
                             float* __restrict__ denom, int Etot, int E)
{
    int e = (int)(blockIdx.x * (size_t)blockDim.x + threadIdx.x);
    if (e >= Etot) return;
    int src, dst; edge_endpoints(ei, e, E, src, dst);
    float w = __expf(evals[e] - emax[dst]);
    evals[e] = w;
    atomicAdd(&denom[dst], w);
}

// Pass C: accum[dst] += h[src] * (w / denom[dst]); one wave per edge
__global__ __launch_bounds__(256) void edge_aggregate(
    const long long* __restrict__ ei, const float* __restrict__ w,
    const float* __restrict__ denom, const float* __restrict__ H,
    float* __restrict__ accum, int Etot, int E)
{
    int e    = (int)((blockIdx.x * (size_t)blockDim.x + threadIdx.x) >> 5);
    int lane = threadIdx.x & 31;
    if (e >= Etot) return;
    int src, dst; edge_endpoints(ei, e, E, src, dst);
    float coef = w[e] / denom[dst];
    float4 h4 = *(const float4*)(H + (size_t)src * HIDC + lane * 4);
    float* o = accum + (size_t)dst * HIDC + lane * 4;
    atomicAdd(o + 0, h4.x * coef);
    atomicAdd(o + 1, h4.y * coef);
    atomicAdd(o + 2, h4.z * coef);
    atomicAdd(o + 3, h4.w * coef);
}

// out = relu(accum + b), in place
__global__ void bias_relu_inplace(float* __restrict__ acc,
                                  const float* __restrict__ b, size_t total)
{
    size_t i = blockIdx.x * (size_t)blockDim.x + threadIdx.x;
    if (i >= total) return;
    float v = acc[i] + b[i & (HIDC - 1)];
    acc[i] = v > 0.f ? v : 0.f;
}

// Heads: factors = h@Wf^T + bf (3), skills = h@Ws^T + bs (7); wave per node.
__global__ __launch_bounds__(256) void heads_kernel(
    const float* __restrict__ H, const float* __restrict__ Wf,
    const float* __restrict__ bf, const float* __restrict__ Ws,
    const float* __restrict__ bs, float* __restrict__ out, int n)
{
    int node = (int)((blockIdx.x * (size_t)blockDim.x + threadIdx.x) >> 5);
    int lane = threadIdx.x & 31;
    if (node >= n) return;
    float4 h4 = *(const float4*)(H + (size_t)node * HIDC + lane * 4);
    float acc[10];
#pragma unroll
    for (int o = 0; o < 10; ++o) {
        const float* wrow = (o < 3) ? (Wf + (size_t)o * HIDC)
                                    : (Ws + (size_t)(o - 3) * HIDC);
        float4 w4 = *(const float4*)(wrow + lane * 4);
        acc[o] = h4.x * w4.x + h4.y * w4.y + h4.z * w4.z + h4.w * w4.w;
    }
#pragma unroll
    for (int o = 0; o < 10; ++o)
#pragma unroll
        for (int off = 16; off > 0; off >>= 1)
            acc[o] += __shfl_xor(acc[o], off, 32);
    if (lane == 0) {
        float* fo = out + (size_t)node * 3;
        float* so = out + (size_t)n * 3 + (size_t)node * 7;
        for (int f = 0; f < 3; ++f) fo[f] = acc[f] + bf[f];
        for (int s = 0; s < 7; ++s) so[s] = acc[3 + s] + bs[s];
    }
}

// ---------------------------------------------------------------------------
// Host orchestration
// ---------------------------------------------------------------------------
static void run_gat_layer(const float* in, const long long* ei,
                          const float* W, const float* avs, const float* avd,
                          const float* bias,
                          float* hbuf, float* accum, float* as, float* ad,
                          float* emax, float* denom, float* evals,
                          int n, int E, hipStream_t stream)
{
    const int Etot = E + n;
    const size_t tot = (size_t)n * HIDC;

    gemm128_wmma_f32<<<(n + 15) / 16, 256, 0, stream>>>(in, W, hbuf, n);
    node_alpha<<<(n + 7) / 8, 256, 0, stream>>>(hbuf, avs, avd, as, ad, n);
    // 'in' is dead after GEMM; accum may alias it (layer 2 reuses buffers)
    init_layer<<<(int)((tot + 255) / 256), 256, 0, stream>>>(accum, emax, denom, n);
    edge_logit_max<<<(Etot + 255) / 256, 256, 0, stream>>>(ei, as, ad, evals, emax, Etot, E);
    edge_exp_sum<<<(Etot + 255) / 256, 256, 0, stream>>>(ei, evals, emax, denom, Etot, E);
    edge_aggregate<<<(Etot + 7) / 8, 256, 0, stream>>>(ei, evals, denom, hbuf, accum, Etot, E);
    bias_relu_inplace<<<(int)((tot + 255) / 256), 256, 0, stream>>>(accum, bias, tot);
}

extern "C" void kernel_launch(void* const* d_in, const int* in_sizes, int n_in,
                              void* d_out, int out_size, void* d_ws, size_t ws_size,
                              hipStream_t stream)
{
    const float*     x   = (const float*)d_in[0];
    const long long* ei  = (const long long*)d_in[1];   // int64 [2,E]
    const float*     W1  = (const float*)d_in[2];
    const float*     as1 = (const float*)d_in[3];
    const float*     ad1 = (const float*)d_in[4];
    const float*     b1  = (const float*)d_in[5];
    const float*     W2  = (const float*)d_in[6];
    const float*     as2 = (const float*)d_in[7];
    const float*     ad2 = (const float*)d_in[8];
    const float*     b2  = (const float*)d_in[9];
    const float*     Wf  = (const float*)d_in[10];
    const float*     bf  = (const float*)d_in[11];
    const float*     Ws  = (const float*)d_in[12];
    const float*     bs  = (const float*)d_in[13];

    const int n = in_sizes[0] / HIDC;      // 100000
    const int E = in_sizes[1] / 2;         // 1600000

    // Workspace layout (floats): h | accum/g | alpha_s | alpha_d | emax | denom | e
    float* ws   = (float*)d_ws;
    size_t nh   = (size_t)n * HIDC;
    float* hbuf = ws;
    float* gbuf = ws + nh;
    float* as   = ws + 2 * nh;
    float* ad   = as + n;
    float* emax = ad + n;
    float* den  = emax + n;
    float* ev   = den + n;                 // (E + n) floats

    // Layer 1: x -> g1 (in gbuf)
    run_gat_layer(x, ei, W1, as1, ad1, b1, hbuf, gbuf, as, ad, emax, den, ev,
                  n, E, stream);
    // Layer 2: g1 -> g2.  GEMM writes hbuf (h1 dead); gbuf is re-zeroed after
    // its contents (g1) were consumed by the GEMM+alpha stage, then reused as
    // the aggregation accumulator.
    run_gat_layer(gbuf, ei, W2, as2, ad2, b2, hbuf, gbuf, as, ad, emax, den, ev,
                  n, E, stream);
    // Heads: g2 -> d_out (factors [n,3] then skills [n,7], flat)
    heads_kernel<<<(n + 7) / 8, 256, 0, stream>>>(gbuf, Wf, bf, Ws, bs,
                                                  (float*)d_out, n);
}